// FlexMaxPool_67173288509590
// MI455X (gfx1250) — compile-verified
//
#include <hip/hip_runtime.h>
#include <cstdint>

#ifndef INFINITY
#define INFINITY __builtin_inff()
#endif

#define CHANNELS 64
#define LANES_PER_EDGE 16   // 64 channels / 4 channels-per-lane
#define BLOCK 256

// native clang vector (accepted by __builtin_nontemporal_load)
typedef long long v2i64 __attribute__((ext_vector_type(2)));

// ---------------------------------------------------------------------------
// Init: segment_max identity is -inf (jax fills empty segments with -inf)
// ---------------------------------------------------------------------------
__global__ void FlexMaxPool_init_kernel(float* __restrict__ out, int n) {
    int i = blockIdx.x * BLOCK + threadIdx.x;
    if (i < n) out[i] = -INFINITY;
}

// ---------------------------------------------------------------------------
// Gather + scatter-max.
//  - 16 lanes cooperate on one edge; each lane owns 4 consecutive channels.
//  - Gathered feature row (256B) is read as one b128 per lane, coalesced.
//  - Index pair (int64 x2 = 16B) read once per edge as a native v2i64, NT
//    hint (streamed exactly once; keep it out of the hot L2 output lines).
//  - Output updated with f32 max atomics -> should lower to
//    global_atomic_max_num_f32 (L2 atomic units, no CAS loop).
// ---------------------------------------------------------------------------
__global__ void FlexMaxPool_scatter_kernel(const float* __restrict__ feats,
                                           const long long* __restrict__ idx,
                                           float* __restrict__ out,
                                           int n_edges) {
    int gid  = blockIdx.x * BLOCK + threadIdx.x;
    int edge = gid >> 4;          // /LANES_PER_EDGE
    int lane = gid & (LANES_PER_EDGE - 1);
    if (edge >= n_edges) return;

    // sparse_indices[edge] = {out_idx, in_idx}, int64 each -> 16B
    const v2i64* ip = reinterpret_cast<const v2i64*>(idx);
    v2i64 e = __builtin_nontemporal_load(&ip[edge]);
    long long out_i = e.x;
    long long in_i  = e.y;

    // coalesced 128-bit gather of this lane's 4 channels
    const float4* src =
        reinterpret_cast<const float4*>(feats + in_i * CHANNELS) + lane;
    float4 v = *src;

    float* dst = out + out_i * CHANNELS + lane * 4;
    atomicMax(dst + 0, v.x);
    atomicMax(dst + 1, v.y);
    atomicMax(dst + 2, v.z);
    atomicMax(dst + 3, v.w);
}

// ---------------------------------------------------------------------------
// kernel_launch
//   d_in[0]: features        float32, N_POINTS*CHANNELS
//   d_in[1]: sparse_indices  int64,   N_EDGES*2
//   d_out  : pooled          float32, N_POINTS*CHANNELS
// ---------------------------------------------------------------------------
extern "C" void kernel_launch(void* const* d_in, const int* in_sizes, int n_in,
                              void* d_out, int out_size, void* d_ws, size_t ws_size,
                              hipStream_t stream) {
    (void)n_in; (void)d_ws; (void)ws_size;

    const float*     feats = (const float*)d_in[0];
    const long long* idx   = (const long long*)d_in[1];
    float*           out   = (float*)d_out;

    int n_edges = in_sizes[1] / 2;

    // 1) fill output with -inf
    int init_blocks = (out_size + BLOCK - 1) / BLOCK;
    FlexMaxPool_init_kernel<<<init_blocks, BLOCK, 0, stream>>>(out, out_size);

    // 2) gather + atomic scatter-max (same stream -> ordered after init)
    long long total_threads = (long long)n_edges * LANES_PER_EDGE;
    int blocks = (int)((total_threads + BLOCK - 1) / BLOCK);
    FlexMaxPool_scatter_kernel<<<blocks, BLOCK, 0, stream>>>(feats, idx, out,
                                                             n_edges);
}